// GNNLayer_28862180229298
// MI455X (gfx1250) — compile-verified
//
#include <hip/hip_runtime.h>

// ---------------------------------------------------------------------------
// GNN layer for MI455X (gfx1250, wave32).
// Dense 64x64 projections via V_WMMA_F32_16X16X4_F32 (exact fp32 WMMA path —
// GEMMs are ~4.9 GFLOP total, negligible vs the L2-atomic-bound scatter, so
// no reason to downconvert precision). W is staged TRANSPOSED in LDS with row
// stride 68 (== 4 mod 64) so each B fragment is one contiguous, bank-conflict-
// free ds_load_b64 (no repacking movs). Edge passes: b64 gathers + f32 atomics.
// ---------------------------------------------------------------------------

typedef float v2f __attribute__((ext_vector_type(2)));
typedef float v8f __attribute__((ext_vector_type(8)));

#define WT_STRIDE 68  // transposed-W LDS row stride; 68 % 64 == 4 -> bank bijection

__device__ __forceinline__ float relu_f(float x) { return fmaxf(x, 0.0f); }

// FUSED=false: Y = X @ W
// FUSED=true : Y = relu( (relu(X) + relu(X2)) @ W )
// One wave computes a 16-row stripe (4 WMMA 16x16 tiles across N=64),
// accumulating K=64 as 16 steps of 16x16x4 fp32 WMMA.
template <bool FUSED>
__global__ __launch_bounds__(256) void gemm64_wmma_kernel(
    const float* __restrict__ X, const float* __restrict__ X2,
    const float* __restrict__ W, float* __restrict__ Y, int M)
{
    __shared__ float Ws[64 * WT_STRIDE];     // Ws[c*WT_STRIDE + k] = W[k][c]
    const int tid = threadIdx.x;
    // Stage W transposed: 256 threads x 1 float4 row-chunk x 4 iters.
    {
        const float4* Wv = (const float4*)W;
        #pragma unroll
        for (int i = 0; i < 4; ++i) {
            const int q = tid + 256 * i;     // float4 index 0..1023
            const float4 v = Wv[q];
            const int r = q >> 4;            // source row (k)   0..63
            const int c = (q & 15) * 4;      // source col base  0..60
            Ws[(c + 0) * WT_STRIDE + r] = v.x;
            Ws[(c + 1) * WT_STRIDE + r] = v.y;
            Ws[(c + 2) * WT_STRIDE + r] = v.z;
            Ws[(c + 3) * WT_STRIDE + r] = v.w;
        }
    }
    __syncthreads();

    const int wave = tid >> 5;
    const int lane = tid & 31;
    const int tile = blockIdx.x * 8 + wave;   // 16-row output tile index
    if (tile * 16 >= M) return;               // wave-uniform guard (EXEC stays full)

    const int r16 = lane & 15;                // row within tile (A) / col (B,C,D)
    const int kh  = lane >> 4;                // lane half selects K pair
    const float* Xr  = X + (size_t)(tile * 16 + r16) * 64;
    const float* X2r = FUSED ? (X2 + (size_t)(tile * 16 + r16) * 64) : nullptr;
    const float* Wt  = Ws + (size_t)r16 * WT_STRIDE + 2 * kh;

    v8f acc0 = {}, acc1 = {}, acc2 = {}, acc3 = {};

    #pragma unroll
    for (int t = 0; t < 16; ++t) {
        const int k0 = 4 * t + 2 * kh;
        // A fragment (16x4 f32): lanes 0-15 carry K={4t,4t+1}, lanes 16-31 K={4t+2,4t+3}
        v2f a;
        if (FUSED) {
            const float2 av = *(const float2*)(Xr + k0);
            const float2 bv = *(const float2*)(X2r + k0);
            a[0] = relu_f(av.x) + relu_f(bv.x);
            a[1] = relu_f(av.y) + relu_f(bv.y);
        } else {
            const float2 av = *(const float2*)(Xr + k0);
            a[0] = av.x; a[1] = av.y;
        }
        // B fragments (4x16 f32): contiguous K-pairs in transposed LDS ->
        // one ds_load_b64 each, conflict-free with WT_STRIDE % 64 == 4.
        const v2f b0 = *(const v2f*)(Wt + 4 * t);
        const v2f b1 = *(const v2f*)(Wt + 4 * t + 16 * WT_STRIDE);
        const v2f b2 = *(const v2f*)(Wt + 4 * t + 32 * WT_STRIDE);
        const v2f b3 = *(const v2f*)(Wt + 4 * t + 48 * WT_STRIDE);

        acc0 = __builtin_amdgcn_wmma_f32_16x16x4_f32(false, a, false, b0, (short)0, acc0, false, false);
        acc1 = __builtin_amdgcn_wmma_f32_16x16x4_f32(false, a, false, b1, (short)0, acc1, false, false);
        acc2 = __builtin_amdgcn_wmma_f32_16x16x4_f32(false, a, false, b2, (short)0, acc2, false, false);
        acc3 = __builtin_amdgcn_wmma_f32_16x16x4_f32(false, a, false, b3, (short)0, acc3, false, false);
    }

    // C/D layout: VGPR r -> row r (lanes 0-15) / row r+8 (lanes 16-31), col = lane&15.
    float* Yt = Y + (size_t)tile * 16 * 64;
    #pragma unroll
    for (int r = 0; r < 8; ++r) {
        const int m = r + 8 * kh;
        float v0 = acc0[r], v1 = acc1[r], v2 = acc2[r], v3 = acc3[r];
        if (FUSED) { v0 = relu_f(v0); v1 = relu_f(v1); v2 = relu_f(v2); v3 = relu_f(v3); }
        float* yr = Yt + (size_t)m * 64;
        yr[ 0 + r16] = v0;
        yr[16 + r16] = v1;
        yr[32 + r16] = v2;
        yr[48 + r16] = v3;
    }
}

// HBS edge pass: acc[dst] += (val[e] * dot(cci[e], a)) * m[src].
// One wave per edge; each lane handles a float2 slice of the 64-float row.
__global__ __launch_bounds__(256) void hbs_edge_kernel(
    const float* __restrict__ m, const int* __restrict__ idx,
    const float* __restrict__ val, const float* __restrict__ cci,
    const float* __restrict__ avec, float* __restrict__ acc, int E)
{
    const int e = blockIdx.x * 8 + (threadIdx.x >> 5);
    if (e >= E) return;
    const int lane = threadIdx.x & 31;
    const int src = idx[e];
    const int dst = idx[E + e];
    const float coeff = val[e] * (cci[3 * e]     * avec[0] +
                                  cci[3 * e + 1] * avec[1] +
                                  cci[3 * e + 2] * avec[2]);
    const float2 v = ((const float2*)(m + (size_t)src * 64))[lane];
    float* d = acc + (size_t)dst * 64 + 2 * lane;
    atomicAdd(d,     coeff * v.x);
    atomicAdd(d + 1, coeff * v.y);
}

// HBNS edge pass over incidence (row in level-0, col in level-1):
//   acc10[row] += v * ms[col];   acc01[col] += v * mt[row]
__global__ __launch_bounds__(256) void hbns_edge_kernel(
    const float* __restrict__ ms, const float* __restrict__ mt,
    const int* __restrict__ idx, const float* __restrict__ val,
    float* __restrict__ acc10, float* __restrict__ acc01, int E)
{
    const int e = blockIdx.x * 8 + (threadIdx.x >> 5);
    if (e >= E) return;
    const int lane = threadIdx.x & 31;
    const int row = idx[e];
    const int col = idx[E + e];
    const float v = val[e];

    const float2 s = ((const float2*)(ms + (size_t)col * 64))[lane];
    float* d0 = acc10 + (size_t)row * 64 + 2 * lane;
    atomicAdd(d0,     v * s.x);
    atomicAdd(d0 + 1, v * s.y);

    const float2 t = ((const float2*)(mt + (size_t)row * 64))[lane];
    float* d1 = acc01 + (size_t)col * 64 + 2 * lane;
    atomicAdd(d1,     v * t.x);
    atomicAdd(d1 + 1, v * t.y);
}

extern "C" void kernel_launch(void* const* d_in, const int* in_sizes, int n_in,
                              void* d_out, int out_size, void* d_ws, size_t ws_size,
                              hipStream_t stream)
{
    const int D = 64;
    const int N0 = in_sizes[0] / D;
    const int N1 = in_sizes[1] / D;
    const int N2 = in_sizes[2] / D;
    const int N3 = in_sizes[3] / D;
    const int N4 = in_sizes[4] / D;
    const int E0  = in_sizes[6];
    const int E1  = in_sizes[8];
    const int NNZ = in_sizes[10];

    const float* x0 = (const float*)d_in[0];
    const float* x1 = (const float*)d_in[1];
    const float* x2 = (const float*)d_in[2];
    const float* x3 = (const float*)d_in[3];
    const float* x4 = (const float*)d_in[4];
    const int*   adj0_idx = (const int*)d_in[5];
    const float* adj0_val = (const float*)d_in[6];
    const int*   adj1_idx = (const int*)d_in[7];
    const float* adj1_val = (const float*)d_in[8];
    const int*   inc_idx  = (const int*)d_in[9];
    const float* inc_val  = (const float*)d_in[10];
    const float* cci00    = (const float*)d_in[11];
    const float* cci11    = (const float*)d_in[12];
    const float* W_hbs0   = (const float*)d_in[13];
    const float* a_hbs0   = (const float*)d_in[14];
    const float* W_hbs1   = (const float*)d_in[15];
    const float* a_hbs1   = (const float*)d_in[16];
    const float* W_hbns_s = (const float*)d_in[17];
    const float* W_hbns_t = (const float*)d_in[18];
    const float* W_aggr0  = (const float*)d_in[19];
    const float* W_aggr1  = (const float*)d_in[20];

    // Workspace layout (floats): accumulators first (zeroed each call), then
    // the projected-message buffers. Total ~205 MB.
    float* ws = (float*)d_ws;
    size_t off = 0;
    float* acc00 = ws + off; off += (size_t)N0 * D;
    float* acc11 = ws + off; off += (size_t)N1 * D;
    float* acc10 = ws + off; off += (size_t)N0 * D;
    float* acc01 = ws + off; off += (size_t)N1 * D;
    const size_t acc_bytes = off * sizeof(float);
    float* m0  = ws + off; off += (size_t)N0 * D;
    float* m1  = ws + off; off += (size_t)N1 * D;
    float* msb = ws + off; off += (size_t)N1 * D;
    float* mtb = ws + off; off += (size_t)N0 * D;

    hipMemsetAsync(ws, 0, acc_bytes, stream);

    const dim3 blk(256);
    auto gemm_blocks = [](int M) { return dim3((unsigned)((M + 127) / 128)); };

    // Stage 1: dense projections (WMMA fp32).
    gemm64_wmma_kernel<false><<<gemm_blocks(N0), blk, 0, stream>>>(x0, nullptr, W_hbs0,  m0,  N0);
    gemm64_wmma_kernel<false><<<gemm_blocks(N1), blk, 0, stream>>>(x1, nullptr, W_hbs1,  m1,  N1);
    gemm64_wmma_kernel<false><<<gemm_blocks(N1), blk, 0, stream>>>(x1, nullptr, W_hbns_s, msb, N1);
    gemm64_wmma_kernel<false><<<gemm_blocks(N0), blk, 0, stream>>>(x0, nullptr, W_hbns_t, mtb, N0);

    // Stage 2: sparse scatter-adds (L2-atomic bound).
    hbs_edge_kernel<<<dim3((unsigned)((E0 + 7) / 8)), blk, 0, stream>>>(
        m0, adj0_idx, adj0_val, cci00, a_hbs0, acc00, E0);
    hbs_edge_kernel<<<dim3((unsigned)((E1 + 7) / 8)), blk, 0, stream>>>(
        m1, adj1_idx, adj1_val, cci11, a_hbs1, acc11, E1);
    hbns_edge_kernel<<<dim3((unsigned)((NNZ + 7) / 8)), blk, 0, stream>>>(
        msb, mtb, inc_idx, inc_val, acc10, acc01, NNZ);

    // Stage 3: fused relu(acc)+relu(acc) -> GEMM -> relu, directly into d_out.
    float* out = (float*)d_out;
    gemm64_wmma_kernel<true><<<gemm_blocks(N0), blk, 0, stream>>>(acc00, acc10, W_aggr0, out, N0);
    gemm64_wmma_kernel<true><<<gemm_blocks(N1), blk, 0, stream>>>(acc11, acc01, W_aggr1,
                                                                  out + (size_t)N0 * D, N1);

    // Pass-through outputs x_2, x_3, x_4.
    size_t o = (size_t)(N0 + N1) * D;
    hipMemcpyAsync(out + o, x2, (size_t)N2 * D * sizeof(float), hipMemcpyDeviceToDevice, stream);
    o += (size_t)N2 * D;
    hipMemcpyAsync(out + o, x3, (size_t)N3 * D * sizeof(float), hipMemcpyDeviceToDevice, stream);
    o += (size_t)N3 * D;
    hipMemcpyAsync(out + o, x4, (size_t)N4 * D * sizeof(float), hipMemcpyDeviceToDevice, stream);
}